// MultiHeadAttention_15736760173191
// MI455X (gfx1250) — compile-verified
//
#include <hip/hip_runtime.h>
#include <hip/hip_bf16.h>
#include <math.h>

// ---------------------------------------------------------------------------
// MultiHeadAttention for MI455X (gfx1250): fp32 WMMA (V_WMMA_F32_16X16X4_F32)
// Outputs (concatenated in d_out): output [B,L,EMBED] then attn [B,H,L,L].
//
// Roofline: attn write (1.07 GB) dominates -> single pass over scores,
// NT stores for streaming outputs so K/V stay resident in the 192 MB L2.
// ---------------------------------------------------------------------------

typedef float v2f __attribute__((ext_vector_type(2)));
typedef float v4f __attribute__((ext_vector_type(4)));
typedef float v8f __attribute__((ext_vector_type(8)));

#define B_SZ   2
#define H_SZ   8
#define L_SZ   4096
#define DK_SZ  64
#define DV_SZ  64
#define EMBED  512

#define QTILE  16         // rows of Q per workgroup
#define NWAVE  8          // 256 threads
#define SROW   4100       // padded LDS row stride for scores (4100 % 64 == 4)
#define QROW   66         // padded LDS row stride for Q tile  (66 % 64 == 2)

__device__ __forceinline__ v8f wmma_f32(v2f a, v2f b, v8f c) {
  return __builtin_amdgcn_wmma_f32_16x16x4_f32(
      /*neg_a=*/false, a, /*neg_b=*/false, b,
      /*c_mod=*/(short)0, c, /*reuse_a=*/false, /*reuse_b=*/false);
}

// ---------------------------------------------------------------------------
// Kernel 1: per (b, h, q-tile of 16): scores -> softmax -> attn out + context
// gridDim.x = B*H*(L/16) = 4096, blockDim.x = 256 (8 waves)
// dynamic LDS: s_scores 16*SROW | s_q 16*QROW | s_ctx 16*64   (~271 KB)
// ---------------------------------------------------------------------------
__global__ void mha_attn_kernel(const float* __restrict__ Q,
                                const float* __restrict__ K,
                                const float* __restrict__ V,
                                float* __restrict__ attn,     // [B,H,L,L]
                                float* __restrict__ ctx_ws) { // [B,L,H,DV]
  constexpr float SCALE = 0.12598815766f;  // 1/sqrt(63), per reference

  extern __shared__ float smem[];
  float* s_scores = smem;                        // 16*SROW
  float* s_q      = s_scores + QTILE * SROW;     // 16*QROW
  float* s_ctx    = s_q + QTILE * QROW;          // 16*64

  const int tid  = threadIdx.x;
  const int wave = tid >> 5;
  const int lane = tid & 31;
  const int ln   = lane & 15;      // row/col index within 16 (fragment role)
  const int hi   = lane >> 4;      // half-select for K-dim of fragments

  const int wg = blockIdx.x;
  const int qt = wg & 255;                 // q-tile index
  const int h  = (wg >> 8) & 7;
  const int b  = wg >> 11;
  const int q0 = qt * QTILE;

  const size_t bh = (size_t)b * H_SZ + h;

  // ---- stage Q tile (16 x 64) into LDS, zero context accumulator ----------
  const float* Qbase = Q + (bh * L_SZ + q0) * DK_SZ;
  for (int i = tid; i < QTILE * DK_SZ; i += 256) {
    int r = i >> 6, c = i & 63;
    s_q[r * QROW + c] = Qbase[r * DK_SZ + c];
  }
  for (int i = tid; i < QTILE * DV_SZ; i += 256) s_ctx[i] = 0.0f;
  __syncthreads();

  // ---- phase 1: S = (Q K^T) * SCALE into LDS -------------------------------
  for (int nt = wave; nt < L_SZ / 16; nt += NWAVE) {
    const int n0 = nt * 16;
    const float* Krow = K + (bh * L_SZ + (n0 + ln)) * DK_SZ;  // key row n0+ln
    const float* qrow = s_q + ln * QROW + 2 * hi;
    v8f acc = {};
    #pragma unroll
    for (int d0 = 0; d0 < DK_SZ; d0 += 4) {
      v2f a  = *(const v2f*)(qrow + d0);
      v2f bf = *(const v2f*)(Krow + d0 + 2 * hi);
      acc = wmma_f32(a, bf, acc);
    }
    const int col = n0 + ln;
    #pragma unroll
    for (int r = 0; r < 8; ++r) {
      int m = hi * 8 + r;
      s_scores[m * SROW + col] = acc[r] * SCALE;
    }
  }
  __syncthreads();

  // ---- phase 2: row softmax; NT-write attn to global, probs back to LDS ---
  #pragma unroll
  for (int rr = 0; rr < 2; ++rr) {
    const int row = wave * 2 + rr;
    float* srow = s_scores + row * SROW;

    float mx = -INFINITY;
    for (int c = lane * 4; c < L_SZ; c += 128) {
      v4f s = *(const v4f*)(srow + c);
      mx = fmaxf(mx, fmaxf(fmaxf(s.x, s.y), fmaxf(s.z, s.w)));
    }
    #pragma unroll
    for (int off = 16; off > 0; off >>= 1)
      mx = fmaxf(mx, __shfl_xor(mx, off, 32));

    float sum = 0.0f;
    for (int c = lane * 4; c < L_SZ; c += 128) {
      v4f s = *(const v4f*)(srow + c);
      v4f p;
      p.x = __expf(s.x - mx); p.y = __expf(s.y - mx);
      p.z = __expf(s.z - mx); p.w = __expf(s.w - mx);
      *(v4f*)(srow + c) = p;
      sum += (p.x + p.y) + (p.z + p.w);
    }
    #pragma unroll
    for (int off = 16; off > 0; off >>= 1)
      sum += __shfl_xor(sum, off, 32);

    const float inv = 1.0f / sum;
    float* arow = attn + ((bh * L_SZ + (q0 + row)) * (size_t)L_SZ);
    for (int c = lane * 4; c < L_SZ; c += 128) {
      v4f p = *(const v4f*)(srow + c);
      p.x *= inv; p.y *= inv; p.z *= inv; p.w *= inv;
      *(v4f*)(srow + c) = p;
      __builtin_nontemporal_store(p, (v4f*)(arow + c));  // streaming: keep L2 for K/V
    }
  }
  __syncthreads();

  // ---- phase 3: context = P (16 x L) * V (L x 64) --------------------------
  // wave -> output col-tile (wave&3)*16, K-half (wave>>2)*2048
  {
    const int n0 = (wave & 3) * 16;
    const int kbeg = (wave >> 2) * (L_SZ / 2);
    const int kend = kbeg + (L_SZ / 2);
    const int n = n0 + ln;
    const float* Vcol = V + bh * (size_t)L_SZ * DV_SZ + n;  // column n of V
    const float* prow = s_scores + ln * SROW + 2 * hi;
    v8f acc = {};
    for (int k0 = kbeg; k0 < kend; k0 += 4) {
      const int kk = k0 + 2 * hi;
      v2f a = *(const v2f*)(prow + k0);
      v2f bf;
      bf.x = Vcol[(size_t)kk * DV_SZ];
      bf.y = Vcol[(size_t)(kk + 1) * DV_SZ];
      acc = wmma_f32(a, bf, acc);
    }
    #pragma unroll
    for (int r = 0; r < 8; ++r) {
      int m = hi * 8 + r;
      atomicAdd(&s_ctx[m * DV_SZ + n], acc[r]);  // ds_add_f32 combine halves
    }
  }
  __syncthreads();

  // ---- write context in transposed layout [b, q, h, v] (RT: reread by FC) --
  for (int i = tid; i < QTILE * DV_SZ; i += 256) {
    int m = i >> 6, v = i & 63;
    ctx_ws[(((size_t)b * L_SZ + (q0 + m)) * H_SZ + h) * DV_SZ + v] = s_ctx[i];
  }
}

// ---------------------------------------------------------------------------
// Kernel 2: output = ctx [8192 x 512] @ fc_w^T   (fc_w is [512 x 512] row-major)
// one wave per 16x16 output tile; 16384 tiles; 8 waves/block -> 2048 blocks
// ---------------------------------------------------------------------------
__global__ void mha_fc_kernel(const float* __restrict__ ctx,
                              const float* __restrict__ fcw,
                              float* __restrict__ out) {
  const int tid  = threadIdx.x;
  const int wave = tid >> 5;
  const int lane = tid & 31;
  const int ln   = lane & 15;
  const int hi   = lane >> 4;

  const int tile = blockIdx.x * NWAVE + wave;        // 0 .. 16383
  const int mt = tile >> 5;                          // 512 row tiles
  const int nt = tile & 31;                          // 32 col tiles
  const int m0 = mt * 16, n0 = nt * 16;

  const float* arow = ctx + (size_t)(m0 + ln) * EMBED + 2 * hi;  // activations
  const float* brow = fcw + (size_t)(n0 + ln) * EMBED + 2 * hi;  // weight row

  v8f acc = {};
  for (int k0 = 0; k0 < EMBED; k0 += 4) {
    v2f a  = *(const v2f*)(arow + k0);
    v2f bf = *(const v2f*)(brow + k0);
    acc = wmma_f32(a, bf, acc);
  }
  #pragma unroll
  for (int r = 0; r < 8; ++r) {
    int m = m0 + hi * 8 + r;
    __builtin_nontemporal_store(acc[r], out + (size_t)m * EMBED + (n0 + ln));
  }
}

// ---------------------------------------------------------------------------
extern "C" void kernel_launch(void* const* d_in, const int* in_sizes, int n_in,
                              void* d_out, int out_size, void* d_ws, size_t ws_size,
                              hipStream_t stream) {
  (void)in_sizes; (void)n_in; (void)out_size; (void)ws_size;

  const float* Q   = (const float*)d_in[0];
  const float* K   = (const float*)d_in[1];
  const float* V   = (const float*)d_in[2];
  const float* fcw = (const float*)d_in[3];

  float* out  = (float*)d_out;                                   // [B,L,EMBED]
  float* attn = out + (size_t)B_SZ * L_SZ * EMBED;               // [B,H,L,L]
  float* ctx  = (float*)d_ws;                                    // [B,L,H,DV]

  const size_t lds_bytes =
      (size_t)(QTILE * SROW + QTILE * QROW + QTILE * DV_SZ) * sizeof(float);

  dim3 grid1(B_SZ * H_SZ * (L_SZ / QTILE));  // 4096
  mha_attn_kernel<<<grid1, 256, lds_bytes, stream>>>(Q, K, V, attn, ctx);

  dim3 grid2((512 * 32) / NWAVE);            // 2048
  mha_fc_kernel<<<grid2, 256, 0, stream>>>(ctx, fcw, out);
}